// PPEncoder_47751446397029
// MI455X (gfx1250) — compile-verified
//
#include <hip/hip_runtime.h>
#include <hip/hip_bf16.h>

// ---------------------------------------------------------------------------
// 2-layer GCN forward for MI455X (gfx1250, wave32).
//   dims: N=100000 nodes, E=3.2M edges, IN=128, H1=32, H2=16
//   GEMMs via V_WMMA_F32_16X16X4_F32 (fp32 WMMA, keeps reference numerics)
//   scatter-add via explicit no-return global_atomic_add_f32 @ device scope
//   (inline asm guarantees the hardware fp atomic, never a CAS loop)
// ---------------------------------------------------------------------------

typedef float v2f __attribute__((ext_vector_type(2)));
typedef float v8f __attribute__((ext_vector_type(8)));

#define THREADS 256
#define WAVES_PER_BLOCK 8   // 256 / 32

// No-return fp32 atomic add, device scope (accumulators are shared across
// WGPs; they live in the 192MB L2 so these resolve at L2 atomic units).
__device__ __forceinline__ void atomic_add_f32_dev(float* addr, float val) {
    asm volatile("global_atomic_add_f32 %0, %1, off scope:SCOPE_DEV"
                 :: "v"(addr), "v"(val)
                 : "memory");
}

// ---------------- elementwise helpers ----------------

__global__ void fill_f32(float* __restrict__ p, float v, int n) {
    int i = blockIdx.x * blockDim.x + threadIdx.x;
    if (i < n) p[i] = v;
}

__global__ void degree_count(const int* __restrict__ dst, float* __restrict__ deg, int nEdges) {
    int e = blockIdx.x * blockDim.x + threadIdx.x;
    if (e < nEdges) {
        atomic_add_f32_dev(&deg[dst[e]], 1.0f);
    }
}

// in-place: deg -> deg^{-1/2}  (deg >= 1 always because of self-loops)
__global__ void inv_sqrt_k(float* __restrict__ deg, int n) {
    int i = blockIdx.x * blockDim.x + threadIdx.x;
    if (i < n) deg[i] = rsqrtf(deg[i]);
}

// ---------------- WMMA GEMM: xw = x[N,128] @ W1[128,32] ----------------
// One wave computes a 16x32 output tile: two 16x16 accumulators, K in steps
// of 4 via V_WMMA_F32_16X16X4_F32.  N = 100000 is an exact multiple of 16.
__global__ __launch_bounds__(THREADS) void gemm1_wmma(
        const float* __restrict__ x, const float* __restrict__ W1,
        float* __restrict__ xw, int nTiles) {
    const int wave = threadIdx.x >> 5;
    const int lane = threadIdx.x & 31;
    const int t = blockIdx.x * WAVES_PER_BLOCK + wave;
    if (t >= nTiles) return;                 // wave-uniform: EXEC stays all-1s

    const int half = lane >> 4;              // 0: K={0,1}, 1: K={2,3}
    const int l    = lane & 15;              // row (A) / col (B,C)
    const int row  = t * 16 + l;
    const float* arow = x + (size_t)row * 128 + half * 2;

    v8f acc0 = {};
    v8f acc1 = {};
    for (int k0 = 0; k0 < 128; k0 += 4) {
        // A fragment (16x4): lane 0-15 -> K=k0+{0,1}, lane 16-31 -> K=k0+{2,3}
        v2f a = *(const v2f*)(arow + k0);
        const int kA = k0 + half * 2;
        // B fragments (4x16), W1 row-major [K=128, N=32]
        v2f b0 = { W1[(size_t)kA * 32 + l],      W1[(size_t)(kA + 1) * 32 + l]      };
        v2f b1 = { W1[(size_t)kA * 32 + 16 + l], W1[(size_t)(kA + 1) * 32 + 16 + l] };
        acc0 = __builtin_amdgcn_wmma_f32_16x16x4_f32(false, a, false, b0,
                                                     (short)0, acc0, false, false);
        acc1 = __builtin_amdgcn_wmma_f32_16x16x4_f32(false, a, false, b1,
                                                     (short)0, acc1, false, false);
    }

    // C/D layout: VGPR v holds row m = v + 8*half, col l
    float* out0 = xw + (size_t)t * 16 * 32;
    #pragma unroll
    for (int v = 0; v < 8; ++v) {
        const int m = v + 8 * half;
        out0[m * 32 + l]      = acc0[v];
        out0[m * 32 + 16 + l] = acc1[v];
    }
}

// ---------------- WMMA GEMM: hw = h[N,32] @ W2[32,16] ----------------
__global__ __launch_bounds__(THREADS) void gemm2_wmma(
        const float* __restrict__ h, const float* __restrict__ W2,
        float* __restrict__ hw, int nTiles) {
    const int wave = threadIdx.x >> 5;
    const int lane = threadIdx.x & 31;
    const int t = blockIdx.x * WAVES_PER_BLOCK + wave;
    if (t >= nTiles) return;

    const int half = lane >> 4;
    const int l    = lane & 15;
    const int row  = t * 16 + l;
    const float* arow = h + (size_t)row * 32 + half * 2;

    v8f acc = {};
    #pragma unroll
    for (int k0 = 0; k0 < 32; k0 += 4) {
        v2f a = *(const v2f*)(arow + k0);
        const int kA = k0 + half * 2;
        v2f b = { W2[(size_t)kA * 16 + l], W2[(size_t)(kA + 1) * 16 + l] };
        acc = __builtin_amdgcn_wmma_f32_16x16x4_f32(false, a, false, b,
                                                    (short)0, acc, false, false);
    }

    float* out0 = hw + (size_t)t * 16 * 16;
    #pragma unroll
    for (int v = 0; v < 8; ++v) {
        const int m = v + 8 * half;
        out0[m * 16 + l] = acc[v];
    }
}

// ---------------- edge scatter: agg[dst] += feat[src] * dinv[s]*dinv[d] ----
// C = channels per node (32 or 16, power of two).  tid = e*C + c, so all
// lanes of a wave share one edge (C=32) or two edges (C=16): the per-edge
// scalar loads coalesce to single cachelines and the feature/atomic accesses
// are contiguous per wave.
template <int C>
__global__ void edge_agg(const int* __restrict__ src, const int* __restrict__ dst,
                         const float* __restrict__ dinv, const float* __restrict__ feat,
                         float* __restrict__ agg, int nEdges) {
    int tid = blockIdx.x * blockDim.x + threadIdx.x;
    int e = tid / C;
    int c = tid % C;
    if (e >= nEdges) return;
    int s = src[e];
    int d = dst[e];
    float w = dinv[s] * dinv[d];
    float v = feat[(size_t)s * C + c] * w;
    atomic_add_f32_dev(&agg[(size_t)d * C + c], v);
}

// h = relu(agg1 + xw * dinv^2 (self-loop) + b1), in place over agg1
__global__ void self_bias_relu(float* __restrict__ agg1, const float* __restrict__ xw,
                               const float* __restrict__ dinv, const float* __restrict__ b1,
                               int nNodes) {
    int tid = blockIdx.x * blockDim.x + threadIdx.x;
    int i = tid >> 5;           // C = 32
    int c = tid & 31;
    if (i >= nNodes) return;
    float di = dinv[i];
    float v = agg1[tid] + xw[tid] * di * di + b1[c];
    agg1[tid] = v > 0.0f ? v : 0.0f;
}

// out = agg2 + hw * dinv^2 (self-loop) + b2
__global__ void self_bias_out(const float* __restrict__ agg2, const float* __restrict__ hw,
                              const float* __restrict__ dinv, const float* __restrict__ b2,
                              float* __restrict__ out, int nNodes) {
    int tid = blockIdx.x * blockDim.x + threadIdx.x;
    int i = tid >> 4;           // C = 16
    int c = tid & 15;
    if (i >= nNodes) return;
    float di = dinv[i];
    out[tid] = agg2[tid] + hw[tid] * di * di + b2[c];
}

// ---------------------------------------------------------------------------

static inline int cdiv(long long a, long long b) { return (int)((a + b - 1) / b); }

extern "C" void kernel_launch(void* const* d_in, const int* in_sizes, int n_in,
                              void* d_out, int out_size, void* d_ws, size_t ws_size,
                              hipStream_t stream) {
    const float* x    = (const float*)d_in[0];   // [N, 128]
    const float* W1   = (const float*)d_in[1];   // [128, 32]
    const float* b1   = (const float*)d_in[2];   // [32]
    const float* W2   = (const float*)d_in[3];   // [32, 16]
    const float* b2   = (const float*)d_in[4];   // [16]
    const int*   eidx = (const int*)d_in[5];     // [2, E]

    const int N = in_sizes[0] / 128;
    const int E = in_sizes[5] / 2;
    const int* src = eidx;
    const int* dst = eidx + E;

    // workspace carve (floats): dinv[N] | xw[N*32] | agg1/h[N*32] | hw[N*16] | agg2[N*16]
    float* ws   = (float*)d_ws;
    float* dinv = ws;
    float* xw   = dinv + (size_t)N;
    float* agg1 = xw   + (size_t)N * 32;   // becomes h after relu
    float* hw   = agg1 + (size_t)N * 32;
    float* agg2 = hw   + (size_t)N * 16;

    float* out = (float*)d_out;            // [N, 16]

    const int nTiles = N / 16;             // N = 100000 is a multiple of 16

    // 1) degrees with self-loop, then dinv = deg^{-1/2}
    fill_f32<<<cdiv(N, THREADS), THREADS, 0, stream>>>(dinv, 1.0f, N);
    degree_count<<<cdiv(E, THREADS), THREADS, 0, stream>>>(dst, dinv, E);
    inv_sqrt_k<<<cdiv(N, THREADS), THREADS, 0, stream>>>(dinv, N);

    // 2) layer 1: xw = x @ W1 (WMMA), edge scatter, self-loop + bias + relu
    gemm1_wmma<<<cdiv(nTiles, WAVES_PER_BLOCK), THREADS, 0, stream>>>(x, W1, xw, nTiles);
    fill_f32<<<cdiv((long long)N * 32, THREADS), THREADS, 0, stream>>>(agg1, 0.0f, N * 32);
    edge_agg<32><<<cdiv((long long)E * 32, THREADS), THREADS, 0, stream>>>(
        src, dst, dinv, xw, agg1, E);
    self_bias_relu<<<cdiv((long long)N * 32, THREADS), THREADS, 0, stream>>>(
        agg1, xw, dinv, b1, N);

    // 3) layer 2: hw = h @ W2 (WMMA), edge scatter, self-loop + bias -> out
    gemm2_wmma<<<cdiv(nTiles, WAVES_PER_BLOCK), THREADS, 0, stream>>>(agg1, W2, hw, nTiles);
    fill_f32<<<cdiv((long long)N * 16, THREADS), THREADS, 0, stream>>>(agg2, 0.0f, N * 16);
    edge_agg<16><<<cdiv((long long)E * 16, THREADS), THREADS, 0, stream>>>(
        src, dst, dinv, hw, agg2, E);
    self_bias_out<<<cdiv((long long)N * 16, THREADS), THREADS, 0, stream>>>(
        agg2, hw, dinv, b2, out, N);
}